// noTF_DetectionConfidenceMap2keypoint_39926015984319
// MI455X (gfx1250) — compile-verified
//
#include <hip/hip_runtime.h>
#include <hip/hip_bf16.h>
#include <stdint.h>

// CDNA5 / gfx1250: wave32, WMMA (not MFMA), async global->LDS DMA.
typedef __attribute__((ext_vector_type(2))) float v2f;
typedef __attribute__((ext_vector_type(8))) float v8f;

#define HH 128
#define WW 128
#define NPIX (HH * WW)          // 16384 elements per (b,c) map
#define STRIDE 132              // padded LDS row stride (dwords): bank = 4*row+col -> conflict-free 16x4 tiles
#define NTHREADS 256            // 8 wave32s
#define NWAVES 8
#define NMAPS (32 * 200)        // B*C = 6400

__global__ __launch_bounds__(NTHREADS)
void hm2kp_kernel(const float* __restrict__ src,
                  float* __restrict__ map_out,
                  float* __restrict__ kp_out,
                  float* __restrict__ zeta_out)
{
    __shared__ float sm[HH * STRIDE];     // 67,584 B: one full padded map
    __shared__ float wred[NWAVES][8];     // per-wave: mn, mx, s0, sx, sy
    __shared__ float bcast[4];

    const int t    = threadIdx.x;
    const int lane = t & 31;
    const int wv   = t >> 5;
    const int mapi = blockIdx.x;          // 0..6399 == (b*200 + c)
    const float* gsrc = src + (size_t)mapi * NPIX;

    // ---------- Phase 1: async DMA the 64KB map into LDS (ASYNCcnt path) ----------
    // 4096 float4 lines / 256 threads = 16 b128 async loads per thread.
    // Generic pointer to __shared__ has the LDS byte offset in its low 32 bits
#pragma unroll
    for (int k = 0; k < 16; ++k) {
        int g4   = t + NTHREADS * k;      // float4 index within map
        int row  = g4 >> 5;               // 32 float4 per 128-float row
        int col4 = g4 & 31;
        uint32_t lds_off = (uint32_t)(uintptr_t)&sm[row * STRIDE + col4 * 4];
        uint64_t gaddr   = (uint64_t)(uintptr_t)(gsrc + (size_t)g4 * 4);
        asm volatile("global_load_async_to_lds_b128 %0, %1, off"
                     :: "v"(lds_off), "v"(gaddr) : "memory");
    }
    asm volatile("s_wait_asynccnt 0" ::: "memory");
    __syncthreads();

    // ---------- Phase 2: per-(b,c) spatial min / max ----------
    float mn = 3.402823466e38f, mx = -3.402823466e38f;
#pragma unroll
    for (int k = 0; k < 64; ++k) {
        int e   = t + NTHREADS * k;
        int row = e >> 7;
        int col = e & 127;
        float v = sm[row * STRIDE + col];
        mn = fminf(mn, v);
        mx = fmaxf(mx, v);
    }
#pragma unroll
    for (int off = 16; off > 0; off >>= 1) {
        mn = fminf(mn, __shfl_xor(mn, off, 32));
        mx = fmaxf(mx, __shfl_xor(mx, off, 32));
    }

    // ---------- Phase 3: raw sums via V_WMMA_F32_16X16X4_F32 ----------
    // Wave wv owns rows [16wv, 16wv+16). A = 16x4 map tile; B = ones (row sums)
    // and B = column index (x-weighted sums). Accumulate C over 32 column chunks.
    // f32 A/B layout (wave32): lanes 0-15 hold K=0,1 in vgpr0/1; lanes 16-31 hold K=2,3.
    v8f acc_s = {0.f,0.f,0.f,0.f,0.f,0.f,0.f,0.f};
    v8f acc_x = {0.f,0.f,0.f,0.f,0.f,0.f,0.f,0.f};
    const int h0   = wv * 16;
    const int arow = h0 + (lane & 15);
    const int kb   = (lane >> 4) * 2;     // 0 or 2
    const float* abase = &sm[arow * STRIDE + kb];
    v2f bones; bones.x = 1.0f; bones.y = 1.0f;
#pragma unroll
    for (int w0 = 0; w0 < WW; w0 += 4) {
        v2f a = *(const v2f*)(abase + w0);           // ds_load_b64, conflict-free
        v2f bx; bx.x = (float)(w0 + kb); bx.y = (float)(w0 + kb + 1);
        acc_s = __builtin_amdgcn_wmma_f32_16x16x4_f32(
                    false, a, false, bones, (short)0, acc_s, false, false);
        acc_x = __builtin_amdgcn_wmma_f32_16x16x4_f32(
                    false, a, false, bx,    (short)0, acc_x, false, false);
    }
    // D layout: vgpr j, lanes 0-15 -> M=j ; lanes 16-31 -> M=8+j (replicated over 16 lanes).
    float s0p = 0.f, sxp = 0.f, syp = 0.f;
    const int mo = h0 + ((lane >> 4) << 3);
#pragma unroll
    for (int j = 0; j < 8; ++j) {
        float rs = acc_s[j];              // row sum of global row (mo + j)
        s0p += rs;
        syp += rs * (float)(mo + j);      // y-weight = row index
        sxp += acc_x[j];                  // already x-weighted row sum
    }
#pragma unroll
    for (int off = 16; off > 0; off >>= 1) {
        s0p += __shfl_xor(s0p, off, 32);
        sxp += __shfl_xor(sxp, off, 32);
        syp += __shfl_xor(syp, off, 32);
    }
    if (lane == 0) {
        wred[wv][0] = mn;
        wred[wv][1] = mx;
        wred[wv][2] = s0p * 0.0625f;      // each value replicated 16x across lanes
        wred[wv][3] = sxp * 0.0625f;
        wred[wv][4] = syp * 0.0625f;
    }
    __syncthreads();

    // ---------- Phase 4: combine waves; emit zeta + keypoint ----------
    if (t == 0) {
        float fmn = wred[0][0], fmx = wred[0][1];
        float S0 = wred[0][2], Sx = wred[0][3], Sy = wred[0][4];
#pragma unroll
        for (int i = 1; i < NWAVES; ++i) {
            fmn = fminf(fmn, wred[i][0]);
            fmx = fmaxf(fmx, wred[i][1]);
            S0 += wred[i][2]; Sx += wred[i][3]; Sy += wred[i][4];
        }
        float inv = 1.0f / fmx;
        // sum_{h,w} w == sum_{h,w} h == 128 * (0+..+127) = 1040384
        const float SUMW = 1040384.0f;
        float zeta = (S0 - 16384.0f * fmn) * inv;  // sum of normalized values
        float kx   = (Sx - SUMW   * fmn) * inv;    // sum x * normalized
        float ky   = (Sy - SUMW   * fmn) * inv;    // sum y * normalized
        zeta_out[mapi]        = zeta;
        kp_out[2 * mapi + 0]  = rintf(kx / zeta);  // RNE like jnp.round
        kp_out[2 * mapi + 1]  = rintf(ky / zeta);
        bcast[0] = fmn;
        bcast[1] = inv;
    }
    __syncthreads();
    const float fmn = bcast[0];
    const float inv = bcast[1];

    // ---------- Phase 5: normalize + stream map_val_all out (single write pass) ----------
    float4* mo4 = (float4*)(map_out + (size_t)mapi * NPIX);
#pragma unroll
    for (int k = 0; k < 16; ++k) {
        int g4   = t + NTHREADS * k;
        int row  = g4 >> 5;
        int col4 = g4 & 31;
        float4 v = *(const float4*)&sm[row * STRIDE + col4 * 4];  // ds_load_b128
        float4 o;
        o.x = (v.x - fmn) * inv;
        o.y = (v.y - fmn) * inv;
        o.z = (v.z - fmn) * inv;
        o.w = (v.w - fmn) * inv;
        mo4[g4] = o;                                              // coalesced b128 store
    }
}

extern "C" void kernel_launch(void* const* d_in, const int* in_sizes, int n_in,
                              void* d_out, int out_size, void* d_ws, size_t ws_size,
                              hipStream_t stream) {
    const float* src = (const float*)d_in[0];   // [32,200,128,128] fp32
    float* map_out  = (float*)d_out;                          // 104,857,600 floats
    float* kp_out   = map_out + (size_t)NMAPS * NPIX;         // 12,800 floats
    float* zeta_out = kp_out + (size_t)NMAPS * 2;             // 6,400 floats
    hm2kp_kernel<<<NMAPS, NTHREADS, 0, stream>>>(src, map_out, kp_out, zeta_out);
}